// GraphAttentionLayer_33397665694526
// MI455X (gfx1250) — compile-verified
//
#include <hip/hip_runtime.h>
#include <math.h>

// Problem constants (from reference)
#define B_    4
#define N_    16384
#define E_    131072
#define HID_  128
#define H_    8
#define D_    16
#define FF_   256
#define BN_   (B_ * N_)

typedef float v2f __attribute__((ext_vector_type(2)));
typedef float v8f __attribute__((ext_vector_type(8)));

// ---------------------------------------------------------------------------
// f32 WMMA helper: D = A(16x4) * B(4x16) + C  (full f32 precision, wave32)
// ---------------------------------------------------------------------------
__device__ __forceinline__ v8f wmma_f32(v2f a, v2f b, v8f c) {
  // 8 args: (neg_a, A, neg_b, B, c_mod, C, reuse_a, reuse_b)
  return __builtin_amdgcn_wmma_f32_16x16x4_f32(false, a, false, b, (short)0, c,
                                               false, false);
}

// A fragment, 16x4 f32 (MxK): lanes 0-15 hold K={k,k+1}, lanes 16-31 K={k+2,k+3}
__device__ __forceinline__ v2f load_a(const float* A, int lda, int row0, int k,
                                      int lane) {
  int m  = row0 + (lane & 15);
  int kk = k + ((lane >> 4) << 1);
  const float* p = A + (size_t)m * lda + kk;
  v2f a;
  a.x = p[0];
  a.y = p[1];
  return a;
}

// float atomic max via integer atomics (works for mixed signs)
__device__ __forceinline__ void atomicMaxF(float* a, float v) {
  if (v >= 0.0f)
    atomicMax((int*)a, __float_as_int(v));
  else
    atomicMin((unsigned int*)a, __float_as_uint(v));
}

// ---------------------------------------------------------------------------
// init: fill node_max with -1e9
// ---------------------------------------------------------------------------
__global__ void fill_f32(float* p, float v, int n) {
  int i = blockIdx.x * blockDim.x + threadIdx.x;
  if (i < n) p[i] = v;
}

// ---------------------------------------------------------------------------
// Weight repack into WMMA B-fragment order:
//   Wp[((ks*ntiles + t)*32 + lane)*2 + {0,1}] = W[kk][n], W[kk+1][n]
//   kk = 4*ks + 2*(lane>>4), n = 16*t + (lane&15)
// Makes every B fragment one coalesced global_load_b64.
// ---------------------------------------------------------------------------
__global__ void repack_w(const float* __restrict__ W, int ncols, int K,
                         float* __restrict__ Wp) {
  int idx = blockIdx.x * blockDim.x + threadIdx.x;
  int ntiles = ncols >> 4;
  int total = (K >> 2) * ntiles * 32;
  if (idx >= total) return;
  int lane = idx & 31;
  int t  = (idx >> 5) % ntiles;
  int ks = (idx >> 5) / ntiles;
  int kk = ks * 4 + ((lane >> 4) << 1);
  int n  = t * 16 + (lane & 15);
  Wp[(size_t)idx * 2 + 0] = W[(size_t)kk * ncols + n];
  Wp[(size_t)idx * 2 + 1] = W[(size_t)(kk + 1) * ncols + n];
}

__device__ __forceinline__ v2f load_bp(const float* Wp, int frag_idx,
                                       int lane) {
  return ((const v2f*)Wp)[((size_t)frag_idx << 5) + lane];
}

// ---------------------------------------------------------------------------
// K1: Q/K/V = h @ {Wq,Wk,Wv}.  grid=(BN/128, 3), block=256 (8 waves).
// ---------------------------------------------------------------------------
__global__ void __launch_bounds__(256)
gemm_qkv(const float* __restrict__ h, const float* __restrict__ Wqp,
         const float* __restrict__ Wkp, const float* __restrict__ Wvp,
         float* __restrict__ Q, float* __restrict__ K, float* __restrict__ V) {
  int lane = threadIdx.x & 31;
  int wave = threadIdx.x >> 5;
  int row0 = (blockIdx.x * 8 + wave) * 16;
  int which = blockIdx.y;
  const float* Wp = (which == 0) ? Wqp : (which == 1) ? Wkp : Wvp;
  float* out = (which == 0) ? Q : (which == 1) ? K : V;

  v8f acc[8] = {};
  for (int ks = 0; ks < HID_ / 4; ++ks) {
    v2f a = load_a(h, HID_, row0, ks * 4, lane);
#pragma unroll
    for (int t = 0; t < 8; ++t) {
      v2f b = load_bp(Wp, ks * 8 + t, lane);
      acc[t] = wmma_f32(a, b, acc[t]);
    }
  }
  int n  = lane & 15;
  int mb = (lane >> 4) * 8;
#pragma unroll
  for (int t = 0; t < 8; ++t)
#pragma unroll
    for (int r = 0; r < 8; ++r)
      out[(size_t)(row0 + mb + r) * HID_ + t * 16 + n] = acc[t][r];
}

// ---------------------------------------------------------------------------
// K2: per-edge logits + atomic node max.  One wave per edge.
// ---------------------------------------------------------------------------
__global__ void edge_logits(const float* __restrict__ Q,
                            const float* __restrict__ K,
                            const int* __restrict__ eidx,
                            const int* __restrict__ nedges,
                            const float* __restrict__ ef,
                            const float* __restrict__ We,
                            float* __restrict__ logits,
                            float* __restrict__ nmax) {
  int gw   = (blockIdx.x * blockDim.x + threadIdx.x) >> 5;
  int lane = threadIdx.x & 31;
  int b = gw >> 17;          // E_ = 2^17
  int e = gw & (E_ - 1);
  if (e >= nedges[b]) return;  // masked edge: contributes nothing

  int src = eidx[(size_t)(b * 2 + 0) * E_ + e];
  int dst = eidx[(size_t)(b * 2 + 1) * E_ + e];

  const float4* qr = (const float4*)(Q + (size_t)(b * N_ + dst) * HID_);
  const float4* kr = (const float4*)(K + (size_t)(b * N_ + src) * HID_);
  float4 q = qr[lane], k = kr[lane];
  float s = q.x * k.x + q.y * k.y + q.z * k.z + q.w * k.w;
  s += __shfl_xor(s, 1);
  s += __shfl_xor(s, 2);   // 4 lanes (one head = 16 floats) reduced

  if ((lane & 3) == 0) {
    int hh = lane >> 2;
    float e0 = ef[(size_t)(b * E_ + e) * 2 + 0];
    float e1 = ef[(size_t)(b * E_ + e) * 2 + 1];
    float lg = s * 0.25f + e0 * We[hh] + e1 * We[H_ + hh];  // scale = D^-0.5
    logits[(size_t)(b * E_ + e) * H_ + hh] = lg;
    atomicMaxF(nmax + (size_t)(b * N_ + dst) * H_ + hh, lg);
  }
}

// ---------------------------------------------------------------------------
// K3: p = exp(logit - max) (masked), denom += p.  One thread per (b,e,h).
// ---------------------------------------------------------------------------
__global__ void edge_p(const int* __restrict__ eidx,
                       const int* __restrict__ nedges,
                       const float* __restrict__ nmax,
                       float* __restrict__ P,   // in: logits, out: p
                       float* __restrict__ den) {
  int idx = blockIdx.x * blockDim.x + threadIdx.x;  // [0, B*E*H)
  int b  = idx >> 20;               // E_*H_ = 2^20
  int e  = (idx >> 3) & (E_ - 1);
  int hh = idx & (H_ - 1);
  float p = 0.0f;
  if (e < nedges[b]) {
    int dst = eidx[(size_t)(b * 2 + 1) * E_ + e];
    float m = fmaxf(nmax[(size_t)(b * N_ + dst) * H_ + hh], -1e9f);
    p = __expf(P[idx] - m);
    unsafeAtomicAdd(den + (size_t)(b * N_ + dst) * H_ + hh, p);
  }
  P[idx] = p;
}

// ---------------------------------------------------------------------------
// K4: out[dst] += (p/denom) * V[src].  One wave per edge, 4 f32 atomics/lane.
// ---------------------------------------------------------------------------
__global__ void edge_scatter(const float* __restrict__ V,
                             const int* __restrict__ eidx,
                             const int* __restrict__ nedges,
                             const float* __restrict__ P,
                             const float* __restrict__ den,
                             float* __restrict__ S) {
  int gw   = (blockIdx.x * blockDim.x + threadIdx.x) >> 5;
  int lane = threadIdx.x & 31;
  int b = gw >> 17;
  int e = gw & (E_ - 1);
  if (e >= nedges[b]) return;

  int src = eidx[(size_t)(b * 2 + 0) * E_ + e];
  int dst = eidx[(size_t)(b * 2 + 1) * E_ + e];
  int hh  = lane >> 2;

  float p  = P[(size_t)(b * E_ + e) * H_ + hh];
  float dn = den[(size_t)(b * N_ + dst) * H_ + hh];
  float alpha = p / fmaxf(dn, 1e-6f);

  float4 v = ((const float4*)(V + (size_t)(b * N_ + src) * HID_))[lane];
  float* o = S + (size_t)(b * N_ + dst) * HID_ + lane * 4;
  unsafeAtomicAdd(o + 0, alpha * v.x);
  unsafeAtomicAdd(o + 1, alpha * v.y);
  unsafeAtomicAdd(o + 2, alpha * v.z);
  unsafeAtomicAdd(o + 3, alpha * v.w);
}

// ---------------------------------------------------------------------------
// Row-wise LayerNorm over HID_=128 cols held in LDS c[16][130].
// One wave; each lane owns 4 columns {lane, lane+32, lane+64, lane+96}.
// ---------------------------------------------------------------------------
__device__ __forceinline__ void ln_rows_store(float (*c)[130], int row0,
                                              const float* g, const float* bl,
                                              float* out, int lane) {
  for (int r = 0; r < 16; ++r) {
    float x0 = c[r][lane], x1 = c[r][lane + 32];
    float x2 = c[r][lane + 64], x3 = c[r][lane + 96];
    float s = x0 + x1 + x2 + x3;
    float q = x0 * x0 + x1 * x1 + x2 * x2 + x3 * x3;
#pragma unroll
    for (int off = 16; off > 0; off >>= 1) {
      s += __shfl_xor(s, off);
      q += __shfl_xor(q, off);
    }
    float mean = s * (1.0f / 128.0f);
    float var  = q * (1.0f / 128.0f) - mean * mean;
    float inv  = rsqrtf(var + 1e-5f);
    float* o = out + (size_t)(row0 + r) * HID_;
    o[lane]      = (x0 - mean) * inv * g[lane]      + bl[lane];
    o[lane + 32] = (x1 - mean) * inv * g[lane + 32] + bl[lane + 32];
    o[lane + 64] = (x2 - mean) * inv * g[lane + 64] + bl[lane + 64];
    o[lane + 96] = (x3 - mean) * inv * g[lane + 96] + bl[lane + 96];
  }
}

// ---------------------------------------------------------------------------
// K5: h1 = LN1(h + S @ Wo + bo).  grid=BN/16, block=32 (one wave).
// ---------------------------------------------------------------------------
__global__ void __launch_bounds__(32)
gemm_wo_ln1(const float* __restrict__ S, const float* __restrict__ Wop,
            const float* __restrict__ bo, const float* __restrict__ h,
            const float* __restrict__ g, const float* __restrict__ bl,
            float* __restrict__ h1) {
  __shared__ float c[16][130];
  int lane = threadIdx.x;
  int row0 = blockIdx.x * 16;

  v8f acc[8] = {};
  for (int ks = 0; ks < HID_ / 4; ++ks) {
    v2f a = load_a(S, HID_, row0, ks * 4, lane);
#pragma unroll
    for (int t = 0; t < 8; ++t) {
      v2f b = load_bp(Wop, ks * 8 + t, lane);
      acc[t] = wmma_f32(a, b, acc[t]);
    }
  }
  int n  = lane & 15;
  int mb = (lane >> 4) * 8;
#pragma unroll
  for (int t = 0; t < 8; ++t)
#pragma unroll
    for (int r = 0; r < 8; ++r) {
      int m = mb + r, col = t * 16 + n;
      c[m][col] = acc[t][r] + bo[col] + h[(size_t)(row0 + m) * HID_ + col];
    }
  __syncthreads();
  ln_rows_store(c, row0, g, bl, h1, lane);
}

// ---------------------------------------------------------------------------
// K6: ff_mid = relu(h1 @ ff1 + b1).  grid=(BN/128, 2), block=256 (8 waves).
// ---------------------------------------------------------------------------
__global__ void __launch_bounds__(256)
gemm_ff1(const float* __restrict__ h1, const float* __restrict__ Wp,
         const float* __restrict__ b1, float* __restrict__ ffm) {
  int lane = threadIdx.x & 31;
  int wave = threadIdx.x >> 5;
  int row0  = (blockIdx.x * 8 + wave) * 16;
  int tile0 = blockIdx.y * 8;   // 16 tiles total across blockIdx.y

  v8f acc[8] = {};
  for (int ks = 0; ks < HID_ / 4; ++ks) {
    v2f a = load_a(h1, HID_, row0, ks * 4, lane);
#pragma unroll
    for (int t = 0; t < 8; ++t) {
      v2f b = load_bp(Wp, ks * 16 + tile0 + t, lane);
      acc[t] = wmma_f32(a, b, acc[t]);
    }
  }
  int n  = lane & 15;
  int mb = (lane >> 4) * 8;
#pragma unroll
  for (int t = 0; t < 8; ++t)
#pragma unroll
    for (int r = 0; r < 8; ++r) {
      int col = (tile0 + t) * 16 + n;
      ffm[(size_t)(row0 + mb + r) * FF_ + col] =
          fmaxf(acc[t][r] + b1[col], 0.0f);
    }
}

// ---------------------------------------------------------------------------
// K7: out = LN2(h1 + ff_mid @ ff2 + b2).  Reads h1 from `out`, overwrites it.
// ---------------------------------------------------------------------------
__global__ void __launch_bounds__(32)
gemm_ff2_ln2(const float* __restrict__ ffm, const float* __restrict__ Wp,
             const float* __restrict__ b2, float* __restrict__ out,
             const float* __restrict__ g, const float* __restrict__ bl) {
  __shared__ float c[16][130];
  int lane = threadIdx.x;
  int row0 = blockIdx.x * 16;

  v8f acc[8] = {};
  for (int ks = 0; ks < FF_ / 4; ++ks) {
    v2f a = load_a(ffm, FF_, row0, ks * 4, lane);
#pragma unroll
    for (int t = 0; t < 8; ++t) {
      v2f b = load_bp(Wp, ks * 8 + t, lane);
      acc[t] = wmma_f32(a, b, acc[t]);
    }
  }
  int n  = lane & 15;
  int mb = (lane >> 4) * 8;
#pragma unroll
  for (int t = 0; t < 8; ++t)
#pragma unroll
    for (int r = 0; r < 8; ++r) {
      int m = mb + r, col = t * 16 + n;
      c[m][col] = acc[t][r] + b2[col] + out[(size_t)(row0 + m) * HID_ + col];
    }
  __syncthreads();
  ln_rows_store(c, row0, g, bl, out, lane);
}

// ---------------------------------------------------------------------------
// Host launcher
// ---------------------------------------------------------------------------
extern "C" void kernel_launch(void* const* d_in, const int* in_sizes, int n_in,
                              void* d_out, int out_size, void* d_ws,
                              size_t ws_size, hipStream_t stream) {
  const float* h    = (const float*)d_in[0];
  const float* ef   = (const float*)d_in[1];
  const int*   eidx = (const int*)d_in[2];
  const int*   ned  = (const int*)d_in[3];
  const float* Wq   = (const float*)d_in[4];
  const float* Wk   = (const float*)d_in[5];
  const float* Wv   = (const float*)d_in[6];
  const float* Wo   = (const float*)d_in[7];
  const float* bo   = (const float*)d_in[8];
  const float* We   = (const float*)d_in[9];
  const float* ln1g = (const float*)d_in[10];
  const float* ln1b = (const float*)d_in[11];
  const float* ln2g = (const float*)d_in[12];
  const float* ln2b = (const float*)d_in[13];
  const float* ff1  = (const float*)d_in[14];
  const float* b1   = (const float*)d_in[15];
  const float* ff2  = (const float*)d_in[16];
  const float* b2   = (const float*)d_in[17];
  float* out = (float*)d_out;

  const size_t MAT    = (size_t)BN_ * HID_;       // 8,388,608 floats
  const size_t PK128  = 32 * 8 * 32 * 2;          // packed 128x128: 16384 floats
  const size_t PKFF1  = 32 * 16 * 32 * 2;         // packed 128x256: 32768
  const size_t PKFF2  = 64 * 8 * 32 * 2;          // packed 256x128: 32768
  float* ws   = (float*)d_ws;
  float* Q    = ws;
  float* K    = ws + MAT;
  float* V    = ws + 2 * MAT;
  float* S    = ws + 3 * MAT;
  float* P    = ws + 4 * MAT;                     // B*E*H
  float* nmax = P + (size_t)B_ * E_ * H_;         // B*N*H
  float* den  = nmax + (size_t)BN_ * H_;          // B*N*H
  float* Wqp  = den + (size_t)BN_ * H_;
  float* Wkp  = Wqp + PK128;
  float* Wvp  = Wkp + PK128;
  float* Wop  = Wvp + PK128;
  float* ff1p = Wop + PK128;
  float* ff2p = ff1p + PKFF1;
  float* ffm  = Q;                                // reuse Q+K region (BN*FF)

  // deterministic re-init every call (graph-replay safe)
  hipMemsetAsync(S, 0, MAT * sizeof(float), stream);
  hipMemsetAsync(den, 0, (size_t)BN_ * H_ * sizeof(float), stream);
  fill_f32<<<(BN_ * H_ + 255) / 256, 256, 0, stream>>>(nmax, -1e9f, BN_ * H_);

  // one-time (per call) weight repacks into WMMA fragment order
  repack_w<<<(PK128 / 2 + 255) / 256, 256, 0, stream>>>(Wq, HID_, HID_, Wqp);
  repack_w<<<(PK128 / 2 + 255) / 256, 256, 0, stream>>>(Wk, HID_, HID_, Wkp);
  repack_w<<<(PK128 / 2 + 255) / 256, 256, 0, stream>>>(Wv, HID_, HID_, Wvp);
  repack_w<<<(PK128 / 2 + 255) / 256, 256, 0, stream>>>(Wo, HID_, HID_, Wop);
  repack_w<<<(PKFF1 / 2 + 255) / 256, 256, 0, stream>>>(ff1, FF_, HID_, ff1p);
  repack_w<<<(PKFF2 / 2 + 255) / 256, 256, 0, stream>>>(ff2, HID_, FF_, ff2p);

  gemm_qkv<<<dim3(BN_ / 128, 3), 256, 0, stream>>>(h, Wqp, Wkp, Wvp, Q, K, V);

  edge_logits<<<(B_ * E_) / 8, 256, 0, stream>>>(Q, K, eidx, ned, ef, We, P,
                                                 nmax);
  edge_p<<<(B_ * E_ * H_) / 256, 256, 0, stream>>>(eidx, ned, nmax, P, den);
  edge_scatter<<<(B_ * E_) / 8, 256, 0, stream>>>(V, eidx, ned, P, den, S);

  gemm_wo_ln1<<<BN_ / 16, 32, 0, stream>>>(S, Wop, bo, h, ln1g, ln1b, out);
  gemm_ff1<<<dim3(BN_ / 128, 2), 256, 0, stream>>>(out, ff1p, b1, ffm);
  gemm_ff2_ln2<<<BN_ / 16, 32, 0, stream>>>(ffm, ff2p, b2, out, ln2g, ln2b);
}